// DGI_8650064134276
// MI455X (gfx1250) — compile-verified
//
#include <hip/hip_runtime.h>

typedef __attribute__((ext_vector_type(2))) float v2f;
typedef __attribute__((ext_vector_type(8))) float v8f;

#define F_DIM 128
#define H_DIM 64

__device__ __forceinline__ void atomAddF(float* p, float v) {
    // no-return fp32 atomic -> global_atomic_add_f32 (STOREcnt path)
    __hip_atomic_fetch_add(p, v, __ATOMIC_RELAXED, __HIP_MEMORY_SCOPE_AGENT);
}

// xt[n,h] = sum_f x[n,f] * W[h,f] + b[h]
// One wave computes a 16(m) x 16(h) tile with V_WMMA_F32_16X16X4_F32,
// looping K=F in steps of 4 (32 WMMAs per wave).
__global__ void __launch_bounds__(256) gemm_wmma_f32(
    const float* __restrict__ x, const float* __restrict__ W,
    const float* __restrict__ bias, float* __restrict__ xt, int N)
{
    const int waveId = (blockIdx.x * blockDim.x + threadIdx.x) >> 5;
    const int lane   = threadIdx.x & 31;
    const int tilesH = H_DIM / 16;            // 4
    const int tileM  = waveId / tilesH;
    const int tileH  = waveId % tilesH;
    if (tileM * 16 >= N) return;              // whole-wave uniform exit

    // A 16x4 f32 layout: lanes 0-15 -> K={0,1}, lanes 16-31 -> K={2,3}; M = lane&15
    // B 4x16 f32 layout: same K split; N(col of W-tile) = lane&15
    const int m   = lane & 15;
    const int kp  = (lane >> 4) << 1;         // 0 or 2
    const float* xrow = x + (size_t)(tileM * 16 + m) * F_DIM;
    const float* wrow = W + (size_t)(tileH * 16 + m) * F_DIM;

    v8f c = {};
    #pragma unroll 8
    for (int k = 0; k < F_DIM; k += 4) {
        v2f a = *(const v2f*)(xrow + k + kp);
        v2f b = *(const v2f*)(wrow + k + kp);
        c = __builtin_amdgcn_wmma_f32_16x16x4_f32(
                /*neg_a=*/false, a, /*neg_b=*/false, b,
                /*c_mod=*/(short)0, c, /*reuse_a=*/false, /*reuse_b=*/false);
    }

    // C/D layout: VGPR v, lanes 0-15 -> M=v, lanes 16-31 -> M=v+8; N = lane&15
    const int hOut  = tileH * 16 + (lane & 15);
    const float bv  = bias[hOut];
    const int mBase = tileM * 16 + ((lane >> 4) << 3);
    #pragma unroll
    for (int v = 0; v < 8; ++v) {
        xt[(size_t)(mBase + v) * H_DIM + hOut] = c[v] + bv;
    }
}

// outH[r,:] += val * xt[c,:]  (16 threads per edge, float4 over H=64)
__global__ void __launch_bounds__(256) spmm_scatter(
    const float* __restrict__ xt, const int* __restrict__ rows,
    const int* __restrict__ cols, const float* __restrict__ vals,
    float* __restrict__ outH, int E)
{
    long long idx = (long long)blockIdx.x * blockDim.x + threadIdx.x;
    long long e = idx >> 4;
    int q = (int)(idx & 15);
    if (e >= E) return;
    const int r = rows[e];
    const int c = cols[e];
    const float v = vals[e];
    const float4 msg = ((const float4*)(xt + (size_t)c * H_DIM))[q];
    float* dst = outH + (size_t)r * H_DIM + q * 4;
    atomAddF(dst + 0, v * msg.x);
    atomAddF(dst + 1, v * msg.y);
    atomAddF(dst + 2, v * msg.z);
    atomAddF(dst + 3, v * msg.w);
}

// In-place PReLU; optionally accumulate per-h column sums (for the readout mean).
// stride is a multiple of 64, so each thread's h-lane (tid%64) is fixed.
__global__ void __launch_bounds__(256) prelu_colsum(
    float* __restrict__ h, const float* __restrict__ prelu_a,
    float* __restrict__ colsum, long long total, int doSum)
{
    const float a = prelu_a[0];
    const long long stride = (long long)gridDim.x * blockDim.x;
    float local = 0.f;
    for (long long i = (long long)blockIdx.x * blockDim.x + threadIdx.x;
         i < total; i += stride) {
        float v = h[i];
        float o = (v >= 0.f) ? v : a * v;
        h[i] = o;
        local += o;
    }
    if (doSum) {
        __shared__ float sm[256];
        sm[threadIdx.x] = local;
        __syncthreads();
        if (threadIdx.x < 64) {
            float t = sm[threadIdx.x] + sm[threadIdx.x + 64] +
                      sm[threadIdx.x + 128] + sm[threadIdx.x + 192];
            atomAddF(colsum + threadIdx.x, t);
        }
    }
}

// s = sigmoid(colsum / N);  t[i] = sum_k W_bil[i,k] * s[k]   (one block, 64 threads)
__global__ void summary_kernel(
    const float* __restrict__ colsum, const float* __restrict__ Wbil,
    float* __restrict__ tbuf, int N)
{
    __shared__ float s_sh[H_DIM];
    const int hh = threadIdx.x;
    float s = colsum[hh] / (float)N;
    s = 1.f / (1.f + expf(-s));
    s_sh[hh] = s;
    __syncthreads();
    float t = 0.f;
    #pragma unroll
    for (int k = 0; k < H_DIM; ++k) t += Wbil[hh * H_DIM + k] * s_sh[k];
    tbuf[hh] = t;
}

// score[n] = dot(Hm[n,:], t) + b   (one wave32 per node, shfl_xor reduction)
__global__ void __launch_bounds__(256) score_kernel(
    const float* __restrict__ Hm, const float* __restrict__ tbuf,
    const float* __restrict__ bbil, float* __restrict__ out, int N)
{
    __shared__ float tsh[H_DIM];
    if (threadIdx.x < H_DIM) tsh[threadIdx.x] = tbuf[threadIdx.x];
    __syncthreads();
    const int wave = threadIdx.x >> 5;
    const int lane = threadIdx.x & 31;
    const int n = blockIdx.x * (blockDim.x >> 5) + wave;
    if (n >= N) return;
    const float* hr = Hm + (size_t)n * H_DIM;
    float acc = hr[lane] * tsh[lane] + hr[lane + 32] * tsh[lane + 32];
    #pragma unroll
    for (int off = 16; off >= 1; off >>= 1)
        acc += __shfl_xor(acc, off, 32);
    if (lane == 0) out[n] = acc + bbil[0];
}

extern "C" void kernel_launch(void* const* d_in, const int* in_sizes, int n_in,
                              void* d_out, int out_size, void* d_ws, size_t ws_size,
                              hipStream_t stream) {
    const float* pos   = (const float*)d_in[0];
    const float* neg   = (const float*)d_in[1];
    const int*   erows = (const int*)  d_in[2];
    const int*   ecols = (const int*)  d_in[3];
    const float* evals = (const float*)d_in[4];
    const float* Wg    = (const float*)d_in[5];
    const float* bg    = (const float*)d_in[6];
    const float* pa    = (const float*)d_in[7];
    const float* Wb    = (const float*)d_in[8];
    const float* bb    = (const float*)d_in[9];

    const int N = in_sizes[0] / F_DIM;   // 100000
    const int E = in_sizes[2];           // 1000000
    float* out = (float*)d_out;

    // workspace: xt (reused pos/neg) | posH | negH | colsum(64) | t(64)
    float* xt     = (float*)d_ws;
    float* posH   = xt   + (size_t)N * H_DIM;
    float* negH   = posH + (size_t)N * H_DIM;
    float* colsum = negH + (size_t)N * H_DIM;
    float* tbuf   = colsum + H_DIM;

    hipMemsetAsync(posH,   0, (size_t)N * H_DIM * sizeof(float), stream);
    hipMemsetAsync(negH,   0, (size_t)N * H_DIM * sizeof(float), stream);
    hipMemsetAsync(colsum, 0, H_DIM * sizeof(float), stream);

    const int tiles      = (N / 16) * (H_DIM / 16);
    const int gemmBlocks = (tiles * 32 + 255) / 256;
    const long long spmmThreads = (long long)E * (H_DIM / 4);
    const int spmmBlocks = (int)((spmmThreads + 255) / 256);
    const long long total = (long long)N * H_DIM;
    const int scoreBlocks = (N + 7) / 8;

    // positive path
    gemm_wmma_f32<<<gemmBlocks, 256, 0, stream>>>(pos, Wg, bg, xt, N);
    spmm_scatter <<<spmmBlocks, 256, 0, stream>>>(xt, erows, ecols, evals, posH, E);
    prelu_colsum <<<2048, 256, 0, stream>>>(posH, pa, colsum, total, 1);
    // negative path (reuses xt after spmm has consumed it — stream ordered)
    gemm_wmma_f32<<<gemmBlocks, 256, 0, stream>>>(neg, Wg, bg, xt, N);
    spmm_scatter <<<spmmBlocks, 256, 0, stream>>>(xt, erows, ecols, evals, negH, E);
    prelu_colsum <<<2048, 256, 0, stream>>>(negH, pa, colsum, total, 0);
    // summary + bilinear vector
    summary_kernel<<<1, 64, 0, stream>>>(colsum, Wb, tbuf, N);
    // scores -> logits [pos | neg]
    score_kernel<<<scoreBlocks, 256, 0, stream>>>(posH, tbuf, bb, out, N);
    score_kernel<<<scoreBlocks, 256, 0, stream>>>(negH, tbuf, bb, out + N, N);
}